// MergedEmbSGD_3410204033833
// MI455X (gfx1250) — compile-verified
//
#include <hip/hip_runtime.h>

typedef float vf4 __attribute__((ext_vector_type(4)));

#define EMB_T 26
#define EMB_N 100000
#define EMB_D 128

__global__ __launch_bounds__(256) void merged_embbag_kernel(
    const float* __restrict__ weights,
    const int* __restrict__ indices,
    const int* __restrict__ offsets,
    float* __restrict__ out,
    int B, int BL)
{
  // 8 waves/block x 2 buffers x 4 rows x 128 floats = 32 KB LDS
  __shared__ float lds[8][2][4][EMB_D];

  const int lane = (int)(threadIdx.x & 31u);
  const int wave = (int)(threadIdx.x >> 5);
  const int t    = (int)blockIdx.y;          // table id: no division anywhere

  const int wavesPerTable = (int)gridDim.x * ((int)blockDim.x >> 5);
  const int wid = (int)blockIdx.x * ((int)blockDim.x >> 5) + wave;

  // Per-table bases: scalar, hoisted out of the bag loop.
  const int*   __restrict__ offs = offsets + (size_t)t * (size_t)B;
  const int*   __restrict__ idxT = indices + (size_t)t * (size_t)BL;
  const float* __restrict__ wT   = weights + (size_t)t * (size_t)EMB_N * EMB_D;
  float*       __restrict__ outT = out     + (size_t)t * (size_t)B * EMB_D;

  struct Bag { const int* ip; int cnt; int c; float* op; };

  auto make_state = [&](int b) {
    Bag s;
    int start = offs[b];
    int end   = (b + 1 < B) ? offs[b + 1] : BL;
    int cnt   = end - start;
    if (cnt < 0) cnt = 0;
    s.ip  = idxT + start;
    s.cnt = cnt;
    s.c   = cnt < 4 ? cnt : 4;
    s.op  = outT + (size_t)b * EMB_D + lane * 4;
    return s;
  };

  // Issue exactly 4 async 16B-per-lane gathers as ONE hardware clause
  // (padded with a safe row so outstanding ASYNCcnt per bag is always 4).
  auto issue_async = [&](const Bag& s, int buf) {
    int r0 = (s.c > 0) ? s.ip[0] : 0;
    int r1 = (s.c > 1) ? s.ip[1] : r0;
    int r2 = (s.c > 2) ? s.ip[2] : r0;
    int r3 = (s.c > 3) ? s.ip[3] : r0;
    unsigned long long a0 = (unsigned long long)(size_t)(wT + (size_t)r0 * EMB_D + lane * 4);
    unsigned long long a1 = (unsigned long long)(size_t)(wT + (size_t)r1 * EMB_D + lane * 4);
    unsigned long long a2 = (unsigned long long)(size_t)(wT + (size_t)r2 * EMB_D + lane * 4);
    unsigned long long a3 = (unsigned long long)(size_t)(wT + (size_t)r3 * EMB_D + lane * 4);
    unsigned l0 = (unsigned)(size_t)&lds[wave][buf][0][lane * 4];
    unsigned l1 = (unsigned)(size_t)&lds[wave][buf][1][lane * 4];
    unsigned l2 = (unsigned)(size_t)&lds[wave][buf][2][lane * 4];
    unsigned l3 = (unsigned)(size_t)&lds[wave][buf][3][lane * 4];
    asm volatile(
        "s_clause 0x3\n\t"
        "global_load_async_to_lds_b128 %0, %4, off\n\t"
        "global_load_async_to_lds_b128 %1, %5, off\n\t"
        "global_load_async_to_lds_b128 %2, %6, off\n\t"
        "global_load_async_to_lds_b128 %3, %7, off"
        :
        : "v"(l0), "v"(l1), "v"(l2), "v"(l3),
          "v"(a0), "v"(a1), "v"(a2), "v"(a3)
        : "memory");
  };

  int parity = 0;
  Bag cur, nxt;
  if (wid < B) {
    cur = make_state(wid);
    issue_async(cur, 0);                      // prologue: prime buffer 0
  }

  for (int b = wid; b < B; b += wavesPerTable) {
    int  bn      = b + wavesPerTable;
    bool hasNext = (bn < B);
    if (hasNext) {
      nxt = make_state(bn);
      issue_async(nxt, parity ^ 1);                     // overlap next gather
      asm volatile("s_wait_asynccnt 0x4" ::: "memory"); // cur bag's 4 done
    } else {
      asm volatile("s_wait_asynccnt 0x0" ::: "memory"); // drain
    }

    vf4 acc = {0.0f, 0.0f, 0.0f, 0.0f};
    if (cur.cnt == 4) {
      // Fast path (reference layout): straight-line 4x b128 LDS reads.
#pragma unroll
      for (int k = 0; k < 4; ++k) {
        vf4 v = *(const vf4*)&lds[wave][parity][k][lane * 4];
        acc += v;
      }
    } else {
      for (int k = 0; k < cur.c; ++k) {
        vf4 v = *(const vf4*)&lds[wave][parity][k][lane * 4];
        acc += v;
      }
      // General-offsets tail (bags with >4 rows): direct vector gathers.
      for (int k = 4; k < cur.cnt; ++k) {
        vf4 v = *(const vf4*)(wT + (size_t)cur.ip[k] * EMB_D + lane * 4);
        acc += v;
      }
    }
    // Output streamed once; keep the 192MB L2 for the weight tables.
    __builtin_nontemporal_store(acc, (vf4*)cur.op);

    cur = nxt;
    parity ^= 1;
  }
}

extern "C" void kernel_launch(void* const* d_in, const int* in_sizes, int n_in,
                              void* d_out, int out_size, void* d_ws, size_t ws_size,
                              hipStream_t stream) {
  const float* weights = (const float*)d_in[0];
  const int*   indices = (const int*)d_in[1];
  const int*   offsets = (const int*)d_in[2];
  float*       out     = (float*)d_out;

  const int B  = in_sizes[2] / EMB_T;  // bags per table
  const int BL = in_sizes[1] / EMB_T;  // indices per table

  // grid.y = table (26); 2048 waves per table, 8 pipelined bags per wave.
  dim3 grid(256, EMB_T), block(256);
  merged_embbag_kernel<<<grid, block, 0, stream>>>(weights, indices, offsets,
                                                   out, B, BL);
}